// FeatureAlign_34677565947937
// MI455X (gfx1250) — compile-verified
//
#include <hip/hip_runtime.h>
#include <cstddef>
#include <cstdint>

typedef __attribute__((ext_vector_type(2))) float v2f;
typedef __attribute__((ext_vector_type(8))) float v8f;

__device__ __forceinline__ v8f wmma4(v2f a, v2f b, v8f c) {
    // V_WMMA_F32_16X16X4_F32: exact fp32, 2048 FLOP/instr
    return __builtin_amdgcn_wmma_f32_16x16x4_f32(false, a, false, b, (short)0, c, false, false);
}

// gfx1250 async global->LDS copy (ASYNCcnt); per-lane scatter, no data VGPRs.
__device__ __forceinline__ void async_copy_b32(uint32_t lds_off, const float* gaddr) {
    asm volatile("global_load_async_to_lds_b32 %0, %1, off"
                 :: "v"(lds_off), "v"((uint64_t)(uintptr_t)gaddr)
                 : "memory");
}
__device__ __forceinline__ void wait_async0() {
    asm volatile("s_wait_asynccnt 0x0" ::: "memory");
}
__device__ __forceinline__ uint32_t lds_off32(const float* p) {
    // flat shared pointer low 32 bits == LDS byte offset
    return (uint32_t)(uintptr_t)p;
}

constexpr int Bn = 2, Cch = 128, H = 128, W = 128, HW = H * W;

// ---------------------------------------------------------------------------
// Permute 3x3 weights [Cout][Cin=128][3][3] -> w2[co][tap*128+ci], zero-pad co
// ---------------------------------------------------------------------------
__global__ __launch_bounds__(256)
void permute_w3x3_kernel(const float* __restrict__ w, float* __restrict__ w2,
                         int Cout, int CoutPad) {
    int i = blockIdx.x * 256 + threadIdx.x;
    int tot = CoutPad * 1152;
    if (i >= tot) return;
    int co = i / 1152;
    int r  = i - co * 1152;
    int tp = r >> 7;
    int ci = r & 127;
    w2[i] = (co < Cout) ? w[(size_t)co * 1152 + ci * 9 + tp] : 0.f;
}

// ---------------------------------------------------------------------------
// off_w prep: fold the exact x2 of (feat_up*2) into weight columns ci>=128
// ---------------------------------------------------------------------------
__global__ __launch_bounds__(256)
void prep_w1x1_kernel(const float* __restrict__ w, float* __restrict__ wout) {
    int i = blockIdx.x * 256 + threadIdx.x;
    if (i >= 128 * 256) return;
    int ci = i & 255;
    wout[i] = w[i] * ((ci >= 128) ? 2.f : 1.f);
}

// ---------------------------------------------------------------------------
// 3x3 conv, pad=1, Cin=128, implicit GEMM via V_WMMA_F32_16X16X4_F32.
// grid (W/32, H, B); each wave: 16 co x 32 pixels, software-pipelined K loop.
// LDS: ci-pair-interleaved halo tile, pair-row stride 48 (bank-clean b64).
// dynamic LDS = 64*3*48*8 = 73728 B. Launch 256 thr (fsm) / 224 thr (om).
// ---------------------------------------------------------------------------
__global__ __launch_bounds__(256)
void conv3x3_wmma_kernel(const float* __restrict__ in, const float* __restrict__ w2,
                         const float* __restrict__ bias, float* __restrict__ out,
                         int CoutStore, int CoutTiles) {
    constexpr int XT = 32;
    constexpr int XPADP = 48;                 // row stride in float2 units
    extern __shared__ float patch[];          // [64 pairs][3][XPADP] x float2

    const int x0 = blockIdx.x * XT;
    const int y  = blockIdx.y;
    const int b  = blockIdx.z;
    const int t  = threadIdx.x;
    const int nt = blockDim.x;

    const float* inb = in + (size_t)b * Cch * HW;
    for (int e = t; e < Cch * 3 * (XT + 2); e += nt) {    // 13056 elements
        int ci = e / (3 * (XT + 2));
        int r  = e - ci * 3 * (XT + 2);
        int ky = r / (XT + 2);
        int xx = r - ky * (XT + 2);
        int gy = y + ky - 1;
        int gx = x0 + xx - 1;
        float* dst = &patch[((((ci >> 1) * 3 + ky) * XPADP + xx) << 1) + (ci & 1)];
        if ((unsigned)gy < (unsigned)H && (unsigned)gx < (unsigned)W)
            async_copy_b32(lds_off32(dst), inb + (size_t)ci * HW + gy * W + gx);
        else
            *dst = 0.f;
    }
    wait_async0();
    __syncthreads();

    const int lane = t & 31;
    const int wave = t >> 5;
    const int nw = nt >> 5;
    const int ln = lane & 15;                 // A: M(co); B/D: N(pixel)
    const int kh = lane >> 4;                 // K-pair select
    const v2f* prow = (const v2f*)patch;

    for (int tile = wave; tile < CoutTiles; tile += nw) {
        const int coBase = tile * 16;
        const float* wrow = w2 + (size_t)(coBase + ln) * 1152;
        v8f acc0 = {}; v8f acc1 = {};

        // flattened K: kk in [0,288); tap = kk>>5, ci0 = (kk&31)*4
        auto ldA = [&](int kk) -> v2f {
            int tap = kk >> 5, ci0 = (kk & 31) << 2;
            return *(const v2f*)(wrow + tap * 128 + ci0 + 2 * kh);
        };
        auto ldB2 = [&](int kk, v2f& o0, v2f& o1) {
            int tap = kk >> 5;
            int ky = (tap * 11) >> 5;         // exact tap/3 for tap<9
            int kx = tap - 3 * ky;
            int p = ((kk & 31) << 1) + kh;
            int bi = p * (3 * XPADP) + ky * XPADP + kx + ln;
            o0 = prow[bi];                    // one ds_load_2addr_b64 pair
            o1 = prow[bi + 16];
        };

        v2f a = ldA(0); v2f b0, b1; ldB2(0, b0, b1);
        for (int kk = 0; kk < 287; ++kk) {    // 1-deep software pipeline
            v2f an = ldA(kk + 1);
            v2f b0n, b1n; ldB2(kk + 1, b0n, b1n);
            acc0 = wmma4(a, b0, acc0);
            acc1 = wmma4(a, b1, acc1);
            a = an; b0 = b0n; b1 = b1n;
        }
        acc0 = wmma4(a, b0, acc0);
        acc1 = wmma4(a, b1, acc1);

        for (int r = 0; r < 8; ++r) {
            int co = coBase + r + 8 * kh;
            if (co < CoutStore) {
                float bv = bias ? bias[co] : 0.f;
                size_t o = ((size_t)(b * CoutStore + co) * H + y) * W + x0;
                out[o + ln]      = acc0[r] + bv;
                out[o + 16 + ln] = acc1[r] + bv;
            }
        }
    }
}

// ---------------------------------------------------------------------------
// 1x1 conv (Cout=128). mode 0: plain NCHW input (Cin ch). mode 1: input is
// concat(feat_arm[128], nearest_up(feat_s)[128]) (x2 folded into weights).
// grid (W/32, H, B), block 256; dynamic LDS = (Cin/2)*48*8 bytes.
// ---------------------------------------------------------------------------
__global__ __launch_bounds__(256)
void conv1x1_wmma_kernel(const float* __restrict__ inA, const float* __restrict__ inS,
                         const float* __restrict__ wk, float* __restrict__ out,
                         int Cin, int mode) {
    constexpr int S2 = 48;                    // pair-row stride (bank-clean)
    extern __shared__ float Bm[];             // [Cin/2][S2] x float2

    const int x0 = blockIdx.x * 32;
    const int y  = blockIdx.y;
    const int b  = blockIdx.z;
    const int t  = threadIdx.x;

    for (int e = t; e < Cin * 32; e += blockDim.x) {
        int ci = e >> 5;
        int n  = e & 31;
        int x  = x0 + n;
        const float* src;
        if (mode == 1) {
            if (ci < 128)
                src = inA + ((size_t)(b * 128 + ci) * H + y) * W + x;
            else
                src = inS + ((size_t)(b * 128 + (ci - 128)) * 64 + (y >> 1)) * 64 + (x >> 1);
        } else {
            src = inA + ((size_t)(b * Cin + ci) * H + y) * W + x;
        }
        float* dst = &Bm[(((ci >> 1) * S2 + n) << 1) + (ci & 1)];
        async_copy_b32(lds_off32(dst), src);
    }
    wait_async0();
    __syncthreads();

    const int lane = t & 31;
    const int wave = t >> 5;
    const int ln = lane & 15;
    const int kh = lane >> 4;
    const int coBase = wave * 16;
    const float* wrow = wk + (size_t)(coBase + ln) * Cin;
    const v2f* Bp = (const v2f*)Bm;

    auto ldA = [&](int kk) -> v2f { return *(const v2f*)(wrow + (kk << 2) + 2 * kh); };
    auto ldB2 = [&](int kk, v2f& o0, v2f& o1) {
        int p = (kk << 1) + kh;
        o0 = Bp[p * S2 + ln];
        o1 = Bp[p * S2 + ln + 16];
    };

    const int KI = Cin >> 2;
    v8f acc0 = {}; v8f acc1 = {};
    v2f a = ldA(0); v2f b0, b1; ldB2(0, b0, b1);
    for (int kk = 0; kk < KI - 1; ++kk) {
        v2f an = ldA(kk + 1);
        v2f b0n, b1n; ldB2(kk + 1, b0n, b1n);
        acc0 = wmma4(a, b0, acc0);
        acc1 = wmma4(a, b1, acc1);
        a = an; b0 = b0n; b1 = b1n;
    }
    acc0 = wmma4(a, b0, acc0);
    acc1 = wmma4(a, b1, acc1);

    for (int r = 0; r < 8; ++r) {
        int co = coBase + r + 8 * kh;
        size_t o = ((size_t)(b * 128 + co) * H + y) * W + x0;
        out[o + ln]      = acc0[r];
        out[o + 16 + ln] = acc1[r];
    }
}

// ---------------------------------------------------------------------------
// Modulated deformable conv (G=8, K=3, pad=1) on feat_up, + bias + ReLU.
// Phase 1: bilinear sample * sigmoid(mask) into LDS, K-pair interleaved.
// Phase 2: 64x16x1152 GEMM via WMMA, software-pipelined.
// grid (W/16, H, B), block 128 (4 waves); dynamic LDS = 73728 B.
// ---------------------------------------------------------------------------
__global__ __launch_bounds__(128)
void dcn_wmma_kernel(const float* __restrict__ fs, const float* __restrict__ om,
                     const float* __restrict__ wd, const float* __restrict__ bias,
                     float* __restrict__ out) {
    extern __shared__ float samp[];           // [576 pairs][16] x float2

    const int x0 = blockIdx.x * 16;
    const int y  = blockIdx.y;
    const int b  = blockIdx.z;
    const int t  = threadIdx.x;

    const float* omb = om + (size_t)b * 216 * HW;
    const float* fsb = fs + (size_t)b * Cch * 64 * 64;

    for (int task = t; task < 16 * 72; task += 128) {
        int m   = task / 72;
        int rem = task - m * 72;
        int g   = rem / 9;
        int tap = rem - g * 9;
        int x   = x0 + m;
        int ch  = g * 9 + tap;
        size_t pix = (size_t)y * W + x;
        float dyv = omb[(size_t)ch * HW + pix];
        float dxv = omb[(size_t)(72 + ch) * HW + pix];
        float ml  = omb[(size_t)(144 + ch) * HW + pix];
        float msk = 1.f / (1.f + expf(-ml));

        int kyy = tap / 3, kxx = tap - 3 * (tap / 3);
        float py = dyv + (float)(kyy + y - 1);
        float px = dxv + (float)(kxx + x - 1);
        float y0f = floorf(py), x0f = floorf(px);
        float wy = py - y0f, wx = px - x0f;
        int iy0 = (int)y0f, ix0 = (int)x0f;

        int yy, xx;
        yy = iy0;     xx = ix0;
        bool v00 = (yy >= 0 && yy < H && xx >= 0 && xx < W);
        int o00 = ((min(max(yy, 0), H - 1)) >> 1) * 64 + ((min(max(xx, 0), W - 1)) >> 1);
        yy = iy0;     xx = ix0 + 1;
        bool v01 = (yy >= 0 && yy < H && xx >= 0 && xx < W);
        int o01 = ((min(max(yy, 0), H - 1)) >> 1) * 64 + ((min(max(xx, 0), W - 1)) >> 1);
        yy = iy0 + 1; xx = ix0;
        bool v10 = (yy >= 0 && yy < H && xx >= 0 && xx < W);
        int o10 = ((min(max(yy, 0), H - 1)) >> 1) * 64 + ((min(max(xx, 0), W - 1)) >> 1);
        yy = iy0 + 1; xx = ix0 + 1;
        bool v11 = (yy >= 0 && yy < H && xx >= 0 && xx < W);
        int o11 = ((min(max(yy, 0), H - 1)) >> 1) * 64 + ((min(max(xx, 0), W - 1)) >> 1);

        float w00 = (1.f - wy) * (1.f - wx) * msk * (v00 ? 1.f : 0.f);
        float w01 = (1.f - wy) * wx         * msk * (v01 ? 1.f : 0.f);
        float w10 = wy * (1.f - wx)         * msk * (v10 ? 1.f : 0.f);
        float w11 = wy * wx                 * msk * (v11 ? 1.f : 0.f);

        const float* fc = fsb + (size_t)(g * 16) * 4096;
        int kbase = (g * 16) * 9 + tap;
        for (int cg = 0; cg < 16; ++cg) {
            const float* f = fc + (size_t)cg * 4096;
            float val = w00 * f[o00] + w01 * f[o01] + w10 * f[o10] + w11 * f[o11];
            int k = kbase + cg * 9;
            samp[(((k >> 1) * 16 + m) << 1) + (k & 1)] = val;
        }
    }
    __syncthreads();

    const int lane = t & 31;
    const int wave = t >> 5;
    const int ln = lane & 15;
    const int kh = lane >> 4;
    const int coBase = wave * 16;
    const float* wrow = wd + (size_t)(coBase + ln) * 1152;
    const v2f* Sp = (const v2f*)samp;

    auto ldA = [&](int kk) -> v2f { return *(const v2f*)(wrow + (kk << 2) + 2 * kh); };
    auto ldB = [&](int kk) -> v2f { return Sp[((kk << 1) + kh) * 16 + ln]; };

    v8f acc = {};
    v2f a = ldA(0); v2f bb = ldB(0);
    for (int kk = 0; kk < 287; ++kk) {
        v2f an = ldA(kk + 1);
        v2f bn = ldB(kk + 1);
        acc = wmma4(a, bb, acc);
        a = an; bb = bn;
    }
    acc = wmma4(a, bb, acc);

    for (int r = 0; r < 8; ++r) {
        int co = coBase + r + 8 * kh;
        float v = acc[r] + bias[co];
        v = fmaxf(v, 0.f);
        out[((size_t)(b * 64 + co) * H + y) * W + x0 + ln] = v;
    }
}

// ---------------------------------------------------------------------------
// BatchNorm batch statistics (training mode, biased var): one block per ch.
// ---------------------------------------------------------------------------
__global__ __launch_bounds__(256)
void bn_stats_kernel(const float* __restrict__ x, float* __restrict__ stats, int Cn) {
    __shared__ float s1[256], s2[256];
    const int c = blockIdx.x;
    float sum = 0.f, sq = 0.f;
    for (int j = threadIdx.x; j < Bn * HW; j += 256) {
        int bb = j >> 14;               // HW = 16384
        int i  = j & (HW - 1);
        float v = x[((size_t)(bb * Cn + c)) * HW + i];
        sum += v; sq += v * v;
    }
    s1[threadIdx.x] = sum; s2[threadIdx.x] = sq;
    __syncthreads();
    for (int s = 128; s > 0; s >>= 1) {
        if (threadIdx.x < s) {
            s1[threadIdx.x] += s1[threadIdx.x + s];
            s2[threadIdx.x] += s2[threadIdx.x + s];
        }
        __syncthreads();
    }
    if (threadIdx.x == 0) {
        float inv = 1.f / (float)(Bn * HW);
        float mean = s1[0] * inv;
        float var  = s2[0] * inv - mean * mean;
        stats[c]      = mean;
        stats[Cn + c] = rsqrtf(var + 1e-5f);
    }
}

// ---------------------------------------------------------------------------
// y = relu((x - m) * istd * g + b) [+ resid]
// ---------------------------------------------------------------------------
__global__ __launch_bounds__(256)
void bn_apply_kernel(const float* __restrict__ x, const float* __restrict__ stats,
                     const float* __restrict__ g, const float* __restrict__ bta,
                     const float* __restrict__ resid, float* __restrict__ out,
                     int Cn, int total) {
    int i = blockIdx.x * 256 + threadIdx.x;
    if (i >= total) return;
    int c = (i >> 14) & (Cn - 1);
    float v = (x[i] - stats[c]) * stats[Cn + c] * g[c] + bta[c];
    v = fmaxf(v, 0.f);
    if (resid) v += resid[i];
    out[i] = v;
}

// ---------------------------------------------------------------------------
extern "C" void kernel_launch(void* const* d_in, const int* in_sizes, int n_in,
                              void* d_out, int out_size, void* d_ws, size_t ws_size,
                              hipStream_t stream) {
    (void)in_sizes; (void)n_in; (void)out_size; (void)ws_size;
    const float* feat_l = (const float*)d_in[0];
    const float* feat_s = (const float*)d_in[1];
    const float* fsm_w  = (const float*)d_in[2];
    const float* fsm_g  = (const float*)d_in[3];
    const float* fsm_b  = (const float*)d_in[4];
    const float* off_w  = (const float*)d_in[5];
    const float* off_g  = (const float*)d_in[6];
    const float* off_b  = (const float*)d_in[7];
    const float* om_w   = (const float*)d_in[8];
    const float* om_b   = (const float*)d_in[9];
    const float* dcn_w  = (const float*)d_in[10];
    const float* dcn_b  = (const float*)d_in[11];
    const float* cat_w  = (const float*)d_in[12];
    const float* cat_g  = (const float*)d_in[13];
    const float* cat_b  = (const float*)d_in[14];

    float* out = (float*)d_out;
    float* ws  = (float*)d_ws;

    const size_t FMAP = (size_t)Bn * Cch * HW;      // 4,194,304 floats
    float* bufA   = ws;                              // fsm pre-BN / cat pre-BN
    float* bufB   = bufA + FMAP;                     // off pre-BN -> offset (in place)
    float* bufC   = bufB + FMAP;                     // off_mask [B][216][H][W]
    float* bufD   = bufC + (size_t)Bn * 216 * HW;    // feat_align [B][64][H][W]
    float* w2fsm  = bufD + (size_t)Bn * 64 * HW;     // permuted fsm weights
    float* w2om   = w2fsm + 128 * 1152;              // permuted+padded om weights
    float* stats0 = w2om + 224 * 1152;
    float* stats1 = stats0 + 256;
    float* stats2 = stats1 + 256;
    float* w1off  = stats2 + 256;                    // off_w with x2 folded in
    float* arm    = out + FMAP;                      // feat_arm -> d_out second half

    // weight prep
    permute_w3x3_kernel<<<(128 * 1152 + 255) / 256, 256, 0, stream>>>(fsm_w, w2fsm, 128, 128);
    permute_w3x3_kernel<<<(224 * 1152 + 255) / 256, 256, 0, stream>>>(om_w,  w2om,  216, 224);
    prep_w1x1_kernel<<<128, 256, 0, stream>>>(off_w, w1off);

    // fsm: conv3x3 -> BN stats -> BN+ReLU (feat_arm straight into d_out)
    conv3x3_wmma_kernel<<<dim3(4, 128, 2), 256, 64 * 3 * 48 * 8, stream>>>(
        feat_l, w2fsm, nullptr, bufA, 128, 8);
    bn_stats_kernel<<<128, 256, 0, stream>>>(bufA, stats0, 128);
    bn_apply_kernel<<<(int)((FMAP + 255) / 256), 256, 0, stream>>>(
        bufA, stats0, fsm_g, fsm_b, nullptr, arm, 128, (int)FMAP);

    // offset: 1x1 conv on concat(arm, up(feat_s)) -> BN stats -> BN+ReLU in place
    conv1x1_wmma_kernel<<<dim3(4, 128, 2), 256, 128 * 48 * 8, stream>>>(
        arm, feat_s, w1off, bufB, 256, 1);
    bn_stats_kernel<<<128, 256, 0, stream>>>(bufB, stats1, 128);
    bn_apply_kernel<<<(int)((FMAP + 255) / 256), 256, 0, stream>>>(
        bufB, stats1, off_g, off_b, nullptr, bufB, 128, (int)FMAP);

    // off_mask: conv3x3 128->216 (+bias), 14 tiles over 7 waves (2 each)
    conv3x3_wmma_kernel<<<dim3(4, 128, 2), 224, 64 * 3 * 48 * 8, stream>>>(
        bufB, w2om, om_b, bufC, 216, 14);

    // modulated deformable conv + bias + ReLU
    dcn_wmma_kernel<<<dim3(8, 128, 2), 128, 576 * 16 * 8, stream>>>(
        feat_s, bufC, dcn_w, dcn_b, bufD);

    // cat: 1x1 conv 64->128 -> BN stats -> BN+ReLU + feat_arm residual -> d_out
    conv1x1_wmma_kernel<<<dim3(4, 128, 2), 256, 32 * 48 * 8, stream>>>(
        bufD, nullptr, cat_w, bufA, 64, 0);
    bn_stats_kernel<<<128, 256, 0, stream>>>(bufA, stats2, 128);
    bn_apply_kernel<<<(int)((FMAP + 255) / 256), 256, 0, stream>>>(
        bufA, stats2, cat_g, cat_b, arm, out, 128, (int)FMAP);
}